// AnmNetwork_4028679323833
// MI455X (gfx1250) — compile-verified
//
#include <hip/hip_runtime.h>
#include <math.h>

// Problem constants (match reference)
#define BATCH 64
#define MDIM  64
#define NMAT  128          // m + l
#define NN    (NMAT * NMAT)
#define MM    (MDIM * MDIM)
#define KIT   10
#define NS_ITERS 14        // Newton-Schulz sign iterations

// Planar complex layout: per batch, [0..NN) = real plane, [NN..2NN) = imag plane.

typedef __attribute__((ext_vector_type(2))) float v2f;
typedef __attribute__((ext_vector_type(8))) float v8f;

// ---------------------------------------------------------------------------
// V_WMMA_F32_16X16X4_F32 wrapper: D(16x16 f32) = A(16x4 f32) * B(4x16 f32) + C
// ---------------------------------------------------------------------------
static __device__ inline v8f wmma4(v2f a, v2f b, v8f c) {
  return __builtin_amdgcn_wmma_f32_16x16x4_f32(false, a, false, b, (short)0, c,
                                               false, false);
}

static __device__ inline v2f load2(const float* p) {
  float2 v = *reinterpret_cast<const float2*>(p);
  v2f r = {v.x, v.y};
  return r;
}

// Deterministic Gaussian noise (stands in for the eigenbasis-aligned random
// eigenvalue perturbation of the reference; splitmix64 -> Box-Muller)
static __device__ inline float gauss_noise(unsigned it, unsigned b, unsigned d) {
  unsigned long long x = ((unsigned long long)(it + 1) << 40) ^
                         ((unsigned long long)(b + 1) << 20) ^
                         (unsigned long long)d ^ 0x9E3779B97F4A7C15ull;
  x ^= x >> 30; x *= 0xBF58476D1CE4E5B9ull;
  x ^= x >> 27; x *= 0x94D049BB133111EBull;
  x ^= x >> 31;
  unsigned u1 = (unsigned)(x & 0xFFFFFFFFu);
  unsigned u2 = (unsigned)(x >> 32);
  float f1 = ((float)u1 + 1.0f) * (1.0f / 4294967296.0f);  // (0,1]
  float f2 = (float)u2 * (1.0f / 4294967296.0f);
  return sqrtf(-2.0f * logf(f1)) * cosf(6.28318530718f * f2);
}

// ---------------------------------------------------------------------------
// Zero workspace state (Theta, Lamda)
// ---------------------------------------------------------------------------
__global__ void zero_kernel(float* __restrict__ p, unsigned long long n) {
  unsigned long long i = (unsigned long long)blockIdx.x * blockDim.x + threadIdx.x;
  unsigned long long stride = (unsigned long long)gridDim.x * blockDim.x;
  for (; i < n; i += stride) p[i] = 0.0f;
}

// ---------------------------------------------------------------------------
// Per-iteration assembly: X, W, Toeplitz T, StackM, Awork = StackM - Lamda/r
// One block per batch element. All matrices planar complex.
// ---------------------------------------------------------------------------
__global__ void build_kernel(const float* __restrict__ Yr, const float* __restrict__ Yi,
                             const float* __restrict__ rho, const float* __restrict__ tau,
                             const float* __restrict__ Theta, const float* __restrict__ Lamda,
                             float* __restrict__ Awork, float* __restrict__ StackM,
                             float* __restrict__ Tbuf, int it)
{
  const int b = blockIdx.x;
  const int tid = threadIdx.x;
  const float r = rho[it];
  const float t = tau[it];
  const float* ThR = Theta + (size_t)b * 2 * NN;
  const float* ThI = ThR + NN;
  const float* LaR = Lamda + (size_t)b * 2 * NN;
  const float* LaI = LaR + NN;
  float* SmR = StackM + (size_t)b * 2 * NN;
  float* SmI = SmR + NN;
  float* AwR = Awork + (size_t)b * 2 * NN;
  float* AwI = AwR + NN;
  float* TbR = Tbuf + (size_t)b * 2 * MM;
  float* TbI = TbR + MM;

  __shared__ float suR[MDIM];
  __shared__ float suI[MDIM];

  // Upper-diagonal sums over top-left m x m blocks of (Lamda + r*Theta)
  if (tid < MDIM) {
    const int k = tid;
    float sR = 0.0f, sI = 0.0f;
    for (int i = 0; i + k < MDIM; ++i) {
      const int e = i * NMAT + (i + k);
      sR += LaR[e] + r * ThR[e];
      sI += LaI[e] + r * ThI[e];
    }
    if (k == 0) sR -= 0.5f * t * (float)MDIM;      // adj[:,0] -= (t/2)*m
    const float nk = 1.0f / (float)(MDIM - k);     // norm[k]
    suR[k] = nk / r * sR;
    suI[k] = nk / r * sI;
  }
  __syncthreads();

  const float invr = 1.0f / r;
  const float xden = 1.0f / (1.0f + 2.0f * r);

  for (int e = tid; e < NN; e += blockDim.x) {
    const int i = e >> 7;         // row
    const int j = e & (NMAT - 1); // col
    float sR, sI;
    if (i < MDIM && j < MDIM) {
      // Toeplitz: T[i][j] = u[j-i] (i<=j) else conj(u[i-j])
      if (i <= j) { sR = suR[j - i]; sI = suI[j - i]; }
      else        { sR = suR[i - j]; sI = -suI[i - j]; }
      TbR[i * MDIM + j] = sR;
      TbI[i * MDIM + j] = sI;
    } else if (i < MDIM) {
      // X quadrant: X[i][j2] = (Y + 2*Lam + 2r*Theta) / (1+2r)
      const int j2 = j - MDIM;
      sR = (Yr[((size_t)b * MDIM + i) * MDIM + j2] + 2.0f * LaR[e] + 2.0f * r * ThR[e]) * xden;
      sI = (Yi[((size_t)b * MDIM + i) * MDIM + j2] + 2.0f * LaI[e] + 2.0f * r * ThI[e]) * xden;
    } else if (j < MDIM) {
      // X^H quadrant: conj(X[j][i-m])
      const int ii = j, j2 = i - MDIM;
      const int ex = ii * NMAT + (MDIM + j2);
      sR =  (Yr[((size_t)b * MDIM + ii) * MDIM + j2] + 2.0f * LaR[ex] + 2.0f * r * ThR[ex]) * xden;
      sI = -(Yi[((size_t)b * MDIM + ii) * MDIM + j2] + 2.0f * LaI[ex] + 2.0f * r * ThI[ex]) * xden;
    } else {
      // W quadrant: Lam/r + Theta - t/(2r)*I
      sR = LaR[e] * invr + ThR[e] - ((i == j) ? t * 0.5f * invr : 0.0f);
      sI = LaI[e] * invr + ThI[e];
    }
    SmR[e] = sR;
    SmI[e] = sI;

    float aR = sR - LaR[e] * invr;
    float aI = sI - LaI[e] * invr;
    if (i == j) aR += gauss_noise((unsigned)it, (unsigned)b, (unsigned)i);
    AwR[e] = aR;
    AwI[e] = aI;
  }
}

// ---------------------------------------------------------------------------
// Frobenius-norm scale: X0 = Awork / ||Awork||_F   (one block per batch)
// ---------------------------------------------------------------------------
__global__ void norm_scale_kernel(const float* __restrict__ Awork,
                                  float* __restrict__ Xcur)
{
  const int b = blockIdx.x;
  const int tid = threadIdx.x;
  __shared__ float red[256];
  const float* A = Awork + (size_t)b * 2 * NN;
  float s = 0.0f;
  for (int e = tid; e < 2 * NN; e += 256) {
    float v = A[e];
    s += v * v;
  }
  red[tid] = s;
  __syncthreads();
  for (int off = 128; off > 0; off >>= 1) {
    if (tid < off) red[tid] += red[tid + off];
    __syncthreads();
  }
  const float inv = rsqrtf(red[0] + 1e-30f);
  float* X = Xcur + (size_t)b * 2 * NN;
  for (int e = tid; e < 2 * NN; e += 256) {
    X[e] = A[e] * inv;
  }
}

// ---------------------------------------------------------------------------
// Batched complex GEMM: C = A * B (128x128 complex per batch, planar).
// One wave32 per 16x16 complex output tile; V_WMMA_F32_16X16X4_F32, K step 4.
// Three accumulators: Cr = (Ar*Br) - (Ai*Bi), Ci = Ar*Bi + Ai*Br; the subtract
// happens once in the epilogue, so the k-loop is pure loads + 4 WMMAs.
// A-fragment 16x4 f32: lanes 0-15 hold K={0,1}, lanes 16-31 K={2,3} (VGPR0/1).
// B-fragment 4x16 f32: VGPR j holds row K = 2*(lane/16)+j, N = lane%16.
// C/D 16x16 f32: VGPR v -> M = v + 8*(lane/16), N = lane%16.
// ---------------------------------------------------------------------------
__global__ void __launch_bounds__(32)
cgemm_kernel(const float* __restrict__ Ab, const float* __restrict__ Bb,
             float* __restrict__ Cb)
{
  const int tx = blockIdx.x;          // col tile
  const int ty = blockIdx.y;          // row tile
  const int b  = blockIdx.z;
  const int lane = threadIdx.x;       // 0..31
  const int hi = lane >> 4;
  const int lo = lane & 15;

  const float* Ar = Ab + (size_t)b * 2 * NN;
  const float* Ai = Ar + NN;
  const float* Br = Bb + (size_t)b * 2 * NN;
  const float* Bi = Br + NN;
  float* Cr = Cb + (size_t)b * 2 * NN;
  float* Ci = Cr + NN;

  const int row = ty * 16 + lo;
  const int col = tx * 16 + lo;

  v8f accRa = {};   // Ar*Br
  v8f accRb = {};   // Ai*Bi
  v8f accI  = {};   // Ar*Bi + Ai*Br

#pragma unroll 4
  for (int kk = 0; kk < NMAT; kk += 4) {
    const int ka = kk + 2 * hi;
    // Planar A: {A[row][ka], A[row][ka+1]} is one contiguous 8B load per plane
    v2f ar = load2(&Ar[row * NMAT + ka]);
    v2f ai = load2(&Ai[row * NMAT + ka]);
    // Planar B: rows ka, ka+1 at this column
    v2f br = {Br[(ka + 0) * NMAT + col], Br[(ka + 1) * NMAT + col]};
    v2f bi = {Bi[(ka + 0) * NMAT + col], Bi[(ka + 1) * NMAT + col]};

    accRa = wmma4(ar, br, accRa);
    accRb = wmma4(ai, bi, accRb);
    accI  = wmma4(ar, bi, accI);
    accI  = wmma4(ai, br, accI);
  }

#pragma unroll
  for (int v = 0; v < 8; ++v) {
    const int rr = ty * 16 + v + 8 * hi;
    Cr[rr * NMAT + col] = accRa[v] - accRb[v];
    Ci[rr * NMAT + col] = accI[v];
  }
}

// ---------------------------------------------------------------------------
// M2 = 1.5*I - 0.5*S1, in place (Newton-Schulz polynomial factor)
// ---------------------------------------------------------------------------
__global__ void m2_kernel(float* __restrict__ S1) {
  const long long total = (long long)BATCH * NN;
  long long idx = (long long)blockIdx.x * blockDim.x + threadIdx.x;
  const long long stride = (long long)gridDim.x * blockDim.x;
  for (; idx < total; idx += stride) {
    const long long bb = idx / NN;
    const int e = (int)(idx % NN);
    const int i = e >> 7, j = e & (NMAT - 1);
    float* R = S1 + bb * 2 * NN;
    float* I = R + NN;
    R[e] = -0.5f * R[e] + ((i == j) ? 1.5f : 0.0f);
    I[e] = -0.5f * I[e];
  }
}

// ---------------------------------------------------------------------------
// Sita = hermitize(0.5*(A + A*S)); Theta <- Sita; Lamda += eta*(Theta - StackM)
// ---------------------------------------------------------------------------
__global__ void finalize_kernel(const float* __restrict__ Awork,
                                const float* __restrict__ P,
                                const float* __restrict__ StackM,
                                float* __restrict__ Theta,
                                float* __restrict__ Lamda,
                                const float* __restrict__ eta, int it)
{
  const float e = eta[it];
  const long long total = (long long)BATCH * NN;
  long long idx = (long long)blockIdx.x * blockDim.x + threadIdx.x;
  const long long stride = (long long)gridDim.x * blockDim.x;
  for (; idx < total; idx += stride) {
    const long long bb = idx / NN;
    const int el = (int)(idx % NN);
    const int i = el >> 7, j = el & (NMAT - 1);
    const int et = j * NMAT + i;   // transposed element
    const float* AwR = Awork + bb * 2 * NN;  const float* AwI = AwR + NN;
    const float* PR  = P     + bb * 2 * NN;  const float* PI  = PR + NN;
    const float* SmR = StackM+ bb * 2 * NN;  const float* SmI = SmR + NN;
    float* ThR = Theta + bb * 2 * NN;        float* ThI = ThR + NN;
    float* LaR = Lamda + bb * 2 * NN;        float* LaI = LaR + NN;

    const float tR = 0.25f * (AwR[el] + PR[el] + AwR[et] + PR[et]);
    const float tI = 0.25f * (AwI[el] + PI[el] - AwI[et] - PI[et]);
    ThR[el] = tR;
    ThI[el] = tI;
    LaR[el] += e * (tR - SmR[el]);
    LaI[el] += e * (tI - SmI[el]);
  }
}

// ---------------------------------------------------------------------------
// Outputs: T as complex128 (interleaved doubles), then uvec
// ---------------------------------------------------------------------------
__global__ void copyT_kernel(const float* __restrict__ Tbuf, double* __restrict__ out) {
  const long long total = (long long)BATCH * MM;
  long long idx = (long long)blockIdx.x * blockDim.x + threadIdx.x;
  const long long stride = (long long)gridDim.x * blockDim.x;
  for (; idx < total; idx += stride) {
    const long long bb = idx / MM;
    const int e = (int)(idx % MM);
    const float* R = Tbuf + bb * 2 * MM;
    const float* I = R + MM;
    out[2 * idx + 0] = (double)R[e];
    out[2 * idx + 1] = (double)I[e];
  }
}

__global__ void uvec_kernel(const float* __restrict__ Tbuf, double* __restrict__ out) {
  const int b = blockIdx.x;
  const int k = threadIdx.x;  // 0..63
  const float* R = Tbuf + (size_t)b * 2 * MM;
  const float* I = R + MM;
  float sR = 0.0f, sI = 0.0f;
  for (int i = k; i < MDIM; ++i) {
    sR += R[i * MDIM + (i - k)];
    sI += I[i * MDIM + (i - k)];
  }
  const float nk = 1.0f / (float)(MDIM - k);
  double* uo = out + 2LL * BATCH * MM;
  uo[2 * ((size_t)b * MDIM + k) + 0] = (double)(sR * nk);
  uo[2 * ((size_t)b * MDIM + k) + 1] = (double)(sI * nk);
}

// ---------------------------------------------------------------------------
// Host orchestration
// ---------------------------------------------------------------------------
extern "C" void kernel_launch(void* const* d_in, const int* in_sizes, int n_in,
                              void* d_out, int out_size, void* d_ws, size_t ws_size,
                              hipStream_t stream) {
  const float* Yr  = (const float*)d_in[0];
  const float* Yi  = (const float*)d_in[1];
  const float* rho = (const float*)d_in[2];
  const float* tau = (const float*)d_in[3];
  const float* eta = (const float*)d_in[4];

  float* ws = (float*)d_ws;
  const size_t MATF = (size_t)BATCH * 2 * NN;   // floats per batched complex matrix set
  float* Theta  = ws + 0 * MATF;
  float* Lamda  = ws + 1 * MATF;
  float* Awork  = ws + 2 * MATF;
  float* StackM = ws + 3 * MATF;
  float* Xa     = ws + 4 * MATF;
  float* S1     = ws + 5 * MATF;
  float* Xb     = ws + 6 * MATF;
  float* Tbuf   = ws + 7 * MATF;

  // Theta and Lamda are adjacent -> single zeroing pass
  zero_kernel<<<512, 256, 0, stream>>>(ws, 2ull * MATF);

  for (int it = 0; it < KIT; ++it) {
    build_kernel<<<BATCH, 256, 0, stream>>>(Yr, Yi, rho, tau, Theta, Lamda,
                                            Awork, StackM, Tbuf, it);
    norm_scale_kernel<<<BATCH, 256, 0, stream>>>(Awork, Xa);

    float* p = Xa;
    float* q = Xb;
    for (int ns = 0; ns < NS_ITERS; ++ns) {
      cgemm_kernel<<<dim3(NMAT / 16, NMAT / 16, BATCH), 32, 0, stream>>>(p, p, S1);
      m2_kernel<<<512, 256, 0, stream>>>(S1);
      cgemm_kernel<<<dim3(NMAT / 16, NMAT / 16, BATCH), 32, 0, stream>>>(p, S1, q);
      float* tmp = p; p = q; q = tmp;
    }
    // P = Awork * sign(Awork)
    cgemm_kernel<<<dim3(NMAT / 16, NMAT / 16, BATCH), 32, 0, stream>>>(Awork, p, S1);
    finalize_kernel<<<512, 256, 0, stream>>>(Awork, S1, StackM, Theta, Lamda, eta, it);
  }

  copyT_kernel<<<512, 256, 0, stream>>>(Tbuf, (double*)d_out);
  uvec_kernel<<<BATCH, MDIM, 0, stream>>>(Tbuf, (double*)d_out);
}